// ModulePopHistory_49357764166206
// MI455X (gfx1250) — compile-verified
//
#include <hip/hip_runtime.h>

// EMA as chained V_WMMA_F32_16X16X4_F32 over a 128-step sliding window.
// f32 precision makes the EMA FIR: 0.8^113 ~ 1e-11 << f32 eps, so each row only
// needs the last <=128 samples before its target index (8 MiB total HBM traffic
// vs 128 MiB for the full scan). Each lane streams its own row's window; A-matrix
// fragments are loaded directly in WMMA register layout via b64 loads (no lane
// exchange needed), all 32 loads batched up front for full memory-level parallelism.
// Exact e0 = x0 boundary restored by closed-form correction when the window clamps to 0.

typedef __attribute__((ext_vector_type(2))) float v2f;
typedef __attribute__((ext_vector_type(8))) float v8f;

#define EMA_ALPHA 0.2f
#define WCHUNKS   8         // window = 8 * 16 = 128 steps; truncation error ~0.8^113

__device__ __forceinline__ float pow08(int e) {     // init-only, e <= 16
    float r = 1.0f;
    for (int i = 0; i < e; ++i) r *= 0.8f;
    return r;
}

__device__ __forceinline__ float powi08(int e) {    // 0.8^e, e in [0, 255]
    float r = 1.0f, b = 0.8f;
#pragma unroll
    for (int i = 0; i < 8; ++i) { if (e & 1) r *= b; b *= b; e >>= 1; }
    return r;
}

template <int IMM>
__device__ __forceinline__ float swzf(float x) {
    return __int_as_float(__builtin_amdgcn_ds_swizzle(__float_as_int(x), IMM));
}

// ds_swizzle group-of-32 immediate: imm = (xor<<10) | (or<<5) | and
#define SWZ_BC15H  0x01F0   // src = (lane & 0x10) | 0x0F : broadcast lane 15 / 31 per half

__global__ __launch_bounds__(128) void ema_wmma_kernel(
    const float* __restrict__ x,      // [B, T] row-major
    const int*   __restrict__ vlen,   // [B]
    float*       __restrict__ out,    // [B]
    int B, int T)
{
    const int lane = threadIdx.x & 31;
    const int wave = (blockIdx.x * blockDim.x + threadIdx.x) >> 5;
    const int base = wave * 16;                 // first of this wave's 16 rows
    if (base >= B) return;                      // wave-uniform guard

    const int n    = lane & 15;                 // N / M-in-half index
    const int half = lane >> 4;

    // ---- per-lane streamed row (row n of the tile): window start, aligned to 16
    const int tln  = vlen[base + n] - 1;
    const int sc_n = max(0, (tln >> 4) - (WCHUNKS - 1));   // start chunk of row n
    // A-frag base: frag c of chunk k = cols 16k + 4c + 2*half {+0,+1}  -> direct b64 loads
    const float* rp = x + (size_t)(base + n) * (size_t)T + (sc_n << 4) + 2 * half;

    // ---- batch ALL window loads up front, already in WMMA A-fragment layout
    v2f A[WCHUNKS][4];
#pragma unroll
    for (int c = 0; c < WCHUNKS; ++c) {
#pragma unroll
        for (int q = 0; q < 4; ++q) {
            A[c][q] = *(const v2f*)(rp + c * 16 + q * 4);   // 8B-aligned global_load_b64
        }
    }

    // ---- capture keys for the 8 rows this half owns in the D layout
    int tl[8], key[8];
#pragma unroll
    for (int r = 0; r < 8; ++r) {
        const int t  = vlen[base + r + 8 * half] - 1;
        const int sc = max(0, (t >> 4) - (WCHUNKS - 1));
        tl[r]  = t;
        key[r] = ((t >> 4) - sc) * 16 + (t & 15);           // hit = chunk*16 + lane
    }

    // ---- B-matrix fragments (constant per lane): frag q covers K rows 4q + 2*half {+0,+1}
    // W[k,j] = (j<=k) ? alpha*0.8^(k-j) : 0   (standard weights; e_{s-1} treated as 0)
    v2f Bf[4];
#pragma unroll
    for (int q = 0; q < 4; ++q) {
        const int kb = 4 * q + 2 * half;
        Bf[q].x = (kb     <= n) ? EMA_ALPHA * pow08(n - kb)     : 0.0f;
        Bf[q].y = (kb + 1 <= n) ? EMA_ALPHA * pow08(n - kb - 1) : 0.0f;
    }
    const float fcarry = pow08(n + 1);          // carry factor 0.8^(N+1)

    float eprev[8], res[8];
#pragma unroll
    for (int r = 0; r < 8; ++r) { eprev[r] = 0.0f; res[r] = 0.0f; }

#pragma unroll
    for (int c = 0; c < WCHUNKS; ++c) {
        // accumulator seeded with carry: C[r] = 0.8^(n+1) * e_prev[row]  (0 for chunk 0)
        v8f acc;
#pragma unroll
        for (int r = 0; r < 8; ++r) acc[r] = (c == 0) ? 0.0f : eprev[r] * fcarry;

        acc = __builtin_amdgcn_wmma_f32_16x16x4_f32(false, A[c][0], false, Bf[0], (short)0, acc, false, false);
        acc = __builtin_amdgcn_wmma_f32_16x16x4_f32(false, A[c][1], false, Bf[1], (short)0, acc, false, false);
        acc = __builtin_amdgcn_wmma_f32_16x16x4_f32(false, A[c][2], false, Bf[2], (short)0, acc, false, false);
        acc = __builtin_amdgcn_wmma_f32_16x16x4_f32(false, A[c][3], false, Bf[3], (short)0, acc, false, false);

        // branch-free result capture + carry update (D[r,15] bcast per half)
        const int cn = c * 16 + n;
#pragma unroll
        for (int r = 0; r < 8; ++r) {
            res[r] = (key[r] == cn) ? acc[r] : res[r];
            if (c < WCHUNKS - 1) eprev[r] = swzf<SWZ_BC15H>(acc[r]);
        }
    }

    // ---- boundary correction (exact e0 = x0 when window starts at t=0) + store
#pragma unroll
    for (int r = 0; r < 8; ++r) {
        if ((tl[r] & 15) == n) {
            const int row = base + r + 8 * half;
            float v = res[r];
            if (tl[r] < 16 * WCHUNKS)              // window clamped to t=0
                v += powi08(tl[r] + 1) * x[(size_t)row * (size_t)T];
            out[row] = v;
        }
    }
}

extern "C" void kernel_launch(void* const* d_in, const int* in_sizes, int n_in,
                              void* d_out, int out_size, void* d_ws, size_t ws_size,
                              hipStream_t stream) {
    const float* x    = (const float*)d_in[0];   // pop_history [B, T] f32
    const int*   vlen = (const int*)d_in[1];     // valid_pop_len [B] i32
    float*       out  = (float*)d_out;           // [B, 1] f32

    const int B = in_sizes[1];
    const int T = in_sizes[0] / B;

    const int wavesTotal    = (B + 15) / 16;     // one wave32 per 16 rows
    const int threads       = 128;               // 4 waves / block
    const int wavesPerBlock = threads / 32;
    const int blocks        = (wavesTotal + wavesPerBlock - 1) / wavesPerBlock;

    hipLaunchKernelGGL(ema_wmma_kernel, dim3(blocks), dim3(threads), 0, stream,
                       x, vlen, out, B, T);
}